// UNO1d_69097433858566
// MI455X (gfx1250) — compile-verified
//
#include <hip/hip_runtime.h>

// ---------------------------------------------------------------------------
// UNO1d on gfx1250: every conv1x1 / truncated rfft / irfft / Fourier resample
// is an fp32 GEMM on V_WMMA_F32_16X16X4_F32.
// GEMM v2: 64x64 block tile, 8 waves, 2 accumulators/wave, float2 K-pair LDS
// layout (ds_load_b64 fragments), uniform fast path with float4 global loads,
// branch-free (select-based) tail path.
// ---------------------------------------------------------------------------

typedef __attribute__((ext_vector_type(2))) float v2f;
typedef __attribute__((ext_vector_type(8))) float v8f;

#define AP 80   // As2 row pitch in float2 (160 floats == 32 mod 64 banks)
#define BP 80   // Bs2 row pitch in float2

__device__ __forceinline__ float gelu_f(float x) {
    const float c0 = 0.7978845608028654f;   // sqrt(2/pi)
    float x3 = x * x * x;
    float t  = tanhf(c0 * (x + 0.044715f * x3));
    return 0.5f * x * (1.0f + t);
}

// C[M,N] = A[M,K] * B[K,N] (+bias[M]) (+add[M,N]) (gelu?)
__global__ __launch_bounds__(256) void gemm_wmma_f32(
    const float* __restrict__ A, int lda, long long sA,
    const float* __restrict__ B, int ldb, long long sB,
    float* __restrict__ C, int ldc, long long sC,
    const float* __restrict__ bias,
    const float* __restrict__ addp, int ldadd, long long sAdd,
    int M, int N, int K, int dogelu)
{
    // LDS: K-pairs as float2 so fragments are single ds_load_b64.
    __shared__ v2f As2[8 * AP];   // [kpair][m], m-tile 64
    __shared__ v2f Bs2[8 * BP];   // [kpair][n], n-tile 64

    const int b = blockIdx.z;
    A += (long long)b * sA;
    B += (long long)b * sB;
    C += (long long)b * sC;
    if (addp) addp += (long long)b * sAdd;

    const int M0 = blockIdx.x * 64;
    const int N0 = blockIdx.y * 64;

    const int t    = threadIdx.x;
    const int lane = t & 31;
    const int wave = t >> 5;            // 0..7
    const int wm   = wave & 3;          // M 16-tile
    const int wn2  = wave >> 2;         // N 32-half (two 16-tiles per wave)
    const int half = lane >> 4;         // K-pair select within fragment
    const int mloc = wm * 16 + (lane & 15);
    const int nl0  = wn2 * 32 + (lane & 15);
    const int nl1  = nl0 + 16;

    // stage loaders
    const int am  = t & 63, akq = t >> 6;    // A: 4 consecutive k per thread
    const int bn  = t & 63, bkq = t >> 6;    // B: 2 k-pairs (4 rows) per thread

    const bool aFast = (M0 + 64 <= M) && ((lda & 3) == 0);
    const bool bFast = (N0 + 64 <= N);

    v8f acc0 = {0.f,0.f,0.f,0.f,0.f,0.f,0.f,0.f};
    v8f acc1 = {0.f,0.f,0.f,0.f,0.f,0.f,0.f,0.f};

    for (int k0 = 0; k0 < K; k0 += 16) {
        const bool ktail = (k0 + 16 > K);

        // ---- stage A tile: As2[kp][m] = (A[m][k0+2kp], A[m][k0+2kp+1]) ----
        if (aFast && !ktail) {
            const float4 q = *reinterpret_cast<const float4*>(
                &A[(long long)(M0 + am) * lda + k0 + (akq << 2)]);
            v2f p0; p0.x = q.x; p0.y = q.y;
            v2f p1; p1.x = q.z; p1.y = q.w;
            As2[((akq << 1) + 0) * AP + am] = p0;
            As2[((akq << 1) + 1) * AP + am] = p1;
        } else {
            const int gm = M0 + am;
            const bool mok = (gm < M);
#pragma unroll
            for (int j = 0; j < 2; ++j) {
                const int kp = (akq << 1) + j;
                const int gk = k0 + (kp << 1);
                bool ok0 = mok && (gk     < K);
                bool ok1 = mok && (gk + 1 < K);
                long long i0 = ok0 ? ((long long)gm * lda + gk    ) : 0;
                long long i1 = ok1 ? ((long long)gm * lda + gk + 1) : 0;
                float v0 = A[i0]; if (!ok0) v0 = 0.f;
                float v1 = A[i1]; if (!ok1) v1 = 0.f;
                v2f p; p.x = v0; p.y = v1;
                As2[kp * AP + am] = p;
            }
        }

        // ---- stage B tile: Bs2[kp][n] = (B[k0+2kp][n], B[k0+2kp+1][n]) ----
        if (bFast && !ktail) {
#pragma unroll
            for (int j = 0; j < 2; ++j) {
                const int kp = (bkq << 1) + j;
                const long long base = (long long)(k0 + (kp << 1)) * ldb + N0 + bn;
                v2f p; p.x = B[base]; p.y = B[base + ldb];
                Bs2[kp * BP + bn] = p;
            }
        } else {
            const int gn = N0 + bn;
            const bool nok = (gn < N);
#pragma unroll
            for (int j = 0; j < 2; ++j) {
                const int kp = (bkq << 1) + j;
                const int gk = k0 + (kp << 1);
                bool ok0 = nok && (gk     < K);
                bool ok1 = nok && (gk + 1 < K);
                long long i0 = ok0 ? ((long long)gk * ldb + gn)       : 0;
                long long i1 = ok1 ? ((long long)(gk + 1) * ldb + gn) : 0;
                float v0 = B[i0]; if (!ok0) v0 = 0.f;
                float v1 = B[i1]; if (!ok1) v1 = 0.f;
                v2f p; p.x = v0; p.y = v1;
                Bs2[kp * BP + bn] = p;
            }
        }

        if (k0 + 16 < K) {  // stage-ahead L2 prefetch
            __builtin_prefetch(&B[(long long)(k0 + 16 + (bkq << 2)) * ldb + N0 + bn], 0, 1);
        }
        __syncthreads();

        // ---- 4 K-steps of 4: each wave does 2 WMMAs per step ----
#pragma unroll
        for (int kk = 0; kk < 16; kk += 4) {
            const int kp = (kk >> 1) + half;   // lane-half selects K pair
            v2f a  = As2[kp * AP + mloc];
            v2f b0 = Bs2[kp * BP + nl0];
            v2f b1 = Bs2[kp * BP + nl1];
            acc0 = __builtin_amdgcn_wmma_f32_16x16x4_f32(
                false, a, false, b0, (short)0, acc0, false, false);
            acc1 = __builtin_amdgcn_wmma_f32_16x16x4_f32(
                false, a, false, b1, (short)0, acc1, false, false);
        }
        __syncthreads();
    }

    // ---- epilogue: row = wm*16 + r + 8*half ----
    const int gmb = M0 + wm * 16 + half * 8;
    const int gn0 = N0 + nl0;
    const int gn1 = N0 + nl1;
#pragma unroll
    for (int r = 0; r < 8; ++r) {
        const int gm = gmb + r;
        if (gm < M) {
            const float bv = bias ? bias[gm] : 0.f;
            if (gn0 < N) {
                float v = acc0[r] + bv;
                if (addp) v += addp[(long long)gm * ldadd + gn0];
                if (dogelu) v = gelu_f(v);
                C[(long long)gm * ldc + gn0] = v;
            }
            if (gn1 < N) {
                float v = acc1[r] + bv;
                if (addp) v += addp[(long long)gm * ldadd + gn1];
                if (dogelu) v = gelu_f(v);
                C[(long long)gm * ldc + gn1] = v;
            }
        }
    }
}

// of[b,o,mm] = sum_i xf[b,i,mm] * (wr + j*wi)[i,o,mm]  (complex, interleaved re/im)
__global__ void spec_mul_kernel(
    const float* __restrict__ xf, const float* __restrict__ wr,
    const float* __restrict__ wi, float* __restrict__ of,
    int ci, int co, int m, int nb)
{
    long long idx = (long long)blockIdx.x * blockDim.x + threadIdx.x;
    long long total = (long long)nb * co * m;
    if (idx >= total) return;
    int mm = (int)(idx % m);
    int o  = (int)((idx / m) % co);
    int bb = (int)(idx / ((long long)m * co));
    const float* xb = xf + (long long)bb * ci * (2 * m);
    float orr = 0.f, oii = 0.f;
    for (int i = 0; i < ci; ++i) {
        float xr = xb[i * 2 * m + 2 * mm];
        float xi = xb[i * 2 * m + 2 * mm + 1];
        long long widx = ((long long)i * co + o) * m + mm;
        float wrv = wr[widx], wiv = wi[widx];
        orr += xr * wrv - xi * wiv;
        oii += xr * wiv + xi * wrv;
    }
    float* ob = of + ((long long)bb * co + o) * (2 * m);
    ob[2 * mm]     = orr;
    ob[2 * mm + 1] = oii;
}

// Forward DFT basis F[N][P]: col 2k = cos(2pi k n/N)/N, col 2k+1 = -sin(..)/N.
__global__ void init_fwd_basis(float* F, int N, int m, int P)
{
    int kp = blockIdx.x * blockDim.x + threadIdx.x;
    int n  = blockIdx.y;
    if (kp >= (P >> 1)) return;
    float* row = F + (long long)n * P;
    if (kp >= m) { row[2 * kp] = 0.f; row[2 * kp + 1] = 0.f; return; }
    long long tt = ((long long)n * kp) % N;
    float ang = 6.283185307179586f * ((float)tt / (float)N);
    float s, c;
    __sincosf(ang, &s, &c);
    float inv = 1.0f / (float)N;
    row[2 * kp]     =  c * inv;
    row[2 * kp + 1] = -s * inv;
}

// Inverse (irfft, norm='forward') basis I[2*mpad][N]:
// k=0: (1,0); k==nyq (output Nyquist): (cos(pi n), 0); else (2cos, -2sin).
__global__ void init_inv_basis(float* I, int N, int m, int mpad, int nyq)
{
    int n = blockIdx.x * blockDim.x + threadIdx.x;
    int k = blockIdx.y;
    if (n >= N || k >= mpad) return;
    float re = 0.f, im = 0.f;
    if (k < m) {
        long long tt = ((long long)n * k) % N;
        float ang = 6.283185307179586f * ((float)tt / (float)N);
        float s, c;
        __sincosf(ang, &s, &c);
        if (k == 0)        { re = 1.f;     im = 0.f;      }
        else if (k == nyq) { re = c;       im = 0.f;      }
        else               { re = 2.f * c; im = -2.f * s; }
    }
    I[(long long)(2 * k) * N + n]     = re;
    I[(long long)(2 * k + 1) * N + n] = im;
}

#define IN(i) ((const float*)d_in[(i)])

extern "C" void kernel_launch(void* const* d_in, const int* in_sizes, int n_in,
                              void* d_out, int out_size, void* d_ws, size_t ws_size,
                              hipStream_t stream)
{
    (void)in_sizes; (void)n_in; (void)out_size; (void)ws_size;
    const int NB = 16;

    // -------- parameters (setup_inputs insertion order) --------
    const float* x       = IN(0);
    const float* lift_w1 = IN(1);  const float* lift_b1 = IN(2);
    const float* lift_w2 = IN(3);  const float* lift_b2 = IN(4);
    const float* horiz_w = IN(5);  const float* horiz_b = IN(6);
    const float* proj_w1 = IN(7);  const float* proj_b1 = IN(8);
    const float* proj_w2 = IN(9);  const float* proj_b2 = IN(10);
    struct LayerP { const float *wr,*wi,*skw,*skb,*msw,*msb,*w1,*b1,*w2,*b2; };
    LayerP LP[5];
    for (int i = 0; i < 5; ++i) {
        int b0 = 11 + 10 * i;
        LP[i] = { IN(b0), IN(b0+1), IN(b0+2), IN(b0+3), IN(b0+4),
                  IN(b0+5), IN(b0+6), IN(b0+7), IN(b0+8), IN(b0+9) };
    }
    const int ci[5] = {128, 64, 128, 128, 192};
    const int co[5] = { 64, 128, 128, 128,  64};
    const int md[5] = { 64,  32,  32,  32,  64};
    const int mh[5] = { 32,  64,  64,  64,  32};
    const int Lin[5]  = {4096, 2048, 2048, 2048, 2048};
    const int Lout[5] = {2048, 2048, 2048, 2048, 4096};

    // -------- workspace layout --------
    float* w = (float*)d_ws;
    size_t off = 0;
    auto alloc = [&](size_t n) -> float* {
        float* p = w + off;
        off += (n + 255) & ~(size_t)255;
        return p;
    };
    float* F4096_64 = alloc((size_t)4096 * 128);
    float* F2048_32 = alloc((size_t)2048 * 64);
    float* F2048_64 = alloc((size_t)2048 * 128);
    float* F4096_R  = alloc((size_t)4096 * 2052);
    float* F2048_R  = alloc((size_t)2048 * 2052);
    float* I2048_64 = alloc((size_t)128 * 2048);
    float* I2048_32 = alloc((size_t)64 * 2048);
    float* I4096_64 = alloc((size_t)128 * 4096);
    float* I2048_R  = alloc((size_t)2052 * 2048);
    float* I4096_R  = alloc((size_t)2052 * 4096);
    float* T0   = alloc((size_t)16 * 256 * 4096);
    float* HA   = alloc((size_t)16 * 128 * 4096);
    float* HB   = alloc((size_t)16 * 128 * 4096);
    float* SKIP0= alloc((size_t)16 * 64 * 2048);
    float* H192 = alloc((size_t)16 * 192 * 2048);
    float* T1   = alloc((size_t)16 * 64 * 4096);
    float* T2   = alloc((size_t)16 * 64 * 4096);
    float* XFR  = alloc((size_t)16 * 64 * 2052);
    float* XFS  = alloc((size_t)16 * 192 * 128);
    float* OF   = alloc((size_t)16 * 128 * 128);
    float* XSK  = alloc((size_t)16 * 128 * 2048);
    float* XMSK = alloc((size_t)16 * 128 * 2048);

    // -------- DFT bases (deterministic, rebuilt every launch) --------
    init_fwd_basis<<<dim3(1, 4096), dim3(64), 0, stream>>>(F4096_64, 4096, 64, 128);
    init_fwd_basis<<<dim3(1, 2048), dim3(64), 0, stream>>>(F2048_32, 2048, 32, 64);
    init_fwd_basis<<<dim3(1, 2048), dim3(64), 0, stream>>>(F2048_64, 2048, 64, 128);
    init_fwd_basis<<<dim3(17, 4096), dim3(64), 0, stream>>>(F4096_R, 4096, 1025, 2052);
    init_fwd_basis<<<dim3(17, 2048), dim3(64), 0, stream>>>(F2048_R, 2048, 1025, 2052);
    init_inv_basis<<<dim3(8, 64),   dim3(256), 0, stream>>>(I2048_64, 2048, 64, 64, -1);
    init_inv_basis<<<dim3(8, 32),   dim3(256), 0, stream>>>(I2048_32, 2048, 32, 32, -1);
    init_inv_basis<<<dim3(16, 64),  dim3(256), 0, stream>>>(I4096_64, 4096, 64, 64, -1);
    init_inv_basis<<<dim3(8, 1026), dim3(256), 0, stream>>>(I2048_R, 2048, 1025, 1026, 1024);
    init_inv_basis<<<dim3(16, 1026),dim3(256), 0, stream>>>(I4096_R, 4096, 1025, 1026, -1);

    auto gemm = [&](const float* A, int lda, long long sA,
                    const float* Bm, int ldb, long long sB,
                    float* Cm, int ldc, long long sC,
                    const float* bias,
                    const float* addp, int ldadd, long long sAdd,
                    int M, int N, int K, int dogelu) {
        dim3 g((M + 63) / 64, (N + 63) / 64, NB);
        gemm_wmma_f32<<<g, dim3(256), 0, stream>>>(
            A, lda, sA, Bm, ldb, sB, Cm, ldc, sC, bias,
            addp, ldadd, sAdd, M, N, K, dogelu);
    };

    // -------- lift --------
    gemm(lift_w1, 2, 0, x, 4096, 2LL * 4096, T0, 4096, 256LL * 4096,
         lift_b1, nullptr, 0, 0, 256, 4096, 2, 1);
    gemm(lift_w2, 256, 0, T0, 4096, 256LL * 4096, HA, 4096, 128LL * 4096,
         lift_b2, nullptr, 0, 0, 128, 4096, 256, 0);

    float* inb [5] = {HA, SKIP0, HB, HB, H192};
    float* spb [5] = {HB, HA, HA, HA, HA};
    float* outb[5] = {SKIP0, HB, HB, H192, HB};

    for (int i = 0; i < 5; ++i) {
        const int CI = ci[i], CO = co[i], M_ = md[i], MH = mh[i];
        const int LI = Lin[i], LO = Lout[i];
        long long sIn  = (i == 4) ? 192LL * 2048 : (long long)CI * LI;
        long long sOut = (i == 3) ? 192LL * 2048 : (long long)CO * LO;
        const int lastf = (i == 4) ? 0 : 1;
        const bool ident = (LI == LO);

        // ---- skip & mskip paths ----
        if (ident) {
            gemm(LP[i].skw, CI, 0, inb[i], LI, sIn, XSK, LO, (long long)CO * LO,
                 LP[i].skb, nullptr, 0, 0, CO, LI, CI, 0);
            gemm(LP[i].msw, CI, 0, inb[i], LI, sIn, XMSK, LO, (long long)CO * LO,
                 LP[i].msb, nullptr, 0, 0, CO, LI, CI, 0);
        } else {
            const float* Fbig = (LI == 4096) ? F4096_R : F2048_R;
            const float* Ibig = (LO == 2048) ? I2048_R : I4096_R;
            // skip: conv -> fwd DFT (1025 modes) -> inverse DFT at LO
            gemm(LP[i].skw, CI, 0, inb[i], LI, sIn, T1, LI, (long long)CO * LI,
                 LP[i].skb, nullptr, 0, 0, CO, LI, CI, 0);
            gemm(T1, LI, (long long)CO * LI, Fbig, 2052, 0, XFR, 2052, (long long)CO * 2052,
                 nullptr, nullptr, 0, 0, CO, 2052, LI, 0);
            gemm(XFR, 2052, (long long)CO * 2052, Ibig, LO, 0, XSK, LO, (long long)CO * LO,
                 nullptr, nullptr, 0, 0, CO, LO, 2052, 0);
            // mskip
            gemm(LP[i].msw, CI, 0, inb[i], LI, sIn, T1, LI, (long long)CO * LI,
                 LP[i].msb, nullptr, 0, 0, CO, LI, CI, 0);
            gemm(T1, LI, (long long)CO * LI, Fbig, 2052, 0, XFR, 2052, (long long)CO * 2052,
                 nullptr, nullptr, 0, 0, CO, 2052, LI, 0);
            gemm(XFR, 2052, (long long)CO * 2052, Ibig, LO, 0, XMSK, LO, (long long)CO * LO,
                 nullptr, nullptr, 0, 0, CO, LO, 2052, 0);
        }

        // ---- spectral conv ----
        const float* Fs = (M_ == 32) ? F2048_32 : ((LI == 4096) ? F4096_64 : F2048_64);
        const float* Is = (M_ == 32) ? I2048_32 : ((LO == 4096) ? I4096_64 : I2048_64);
        const int twoM = 2 * M_;
        gemm(inb[i], LI, sIn, Fs, twoM, 0, XFS, twoM, (long long)CI * twoM,
             nullptr, nullptr, 0, 0, CI, twoM, LI, 0);
        {
            long long tot = (long long)NB * CO * M_;
            spec_mul_kernel<<<dim3((unsigned)((tot + 255) / 256)), dim3(256), 0, stream>>>(
                XFS, LP[i].wr, LP[i].wi, OF, CI, CO, M_, NB);
        }
        // irfft + skip add (+gelu unless last)
        gemm(OF, twoM, (long long)CO * twoM, Is, LO, 0, spb[i], LO, (long long)CO * LO,
             nullptr, XSK, LO, (long long)CO * LO, CO, LO, twoM, lastf);

        // ---- channel MLP ----
        gemm(LP[i].w1, CO, 0, spb[i], LO, (long long)CO * LO, T2, LO, (long long)MH * LO,
             LP[i].b1, nullptr, 0, 0, MH, LO, CO, 1);
        gemm(LP[i].w2, MH, 0, T2, LO, (long long)MH * LO, outb[i], LO, sOut,
             LP[i].b2, XMSK, LO, (long long)CO * LO, CO, LO, MH, lastf);

        // after layer 3: horizontal skip conv (resample 2048->2048 = identity)
        // into channels 128..191 of the concat buffer for layer 4
        if (i == 3) {
            gemm(horiz_w, 64, 0, SKIP0, 2048, 64LL * 2048,
                 H192 + (size_t)128 * 2048, 2048, 192LL * 2048,
                 horiz_b, nullptr, 0, 0, 64, 2048, 64, 0);
        }
    }

    // -------- projection head --------
    gemm(proj_w1, 64, 0, HB, 4096, 64LL * 4096, T0, 4096, 256LL * 4096,
         proj_b1, nullptr, 0, 0, 256, 4096, 64, 1);
    gemm(proj_w2, 256, 0, T0, 4096, 256LL * 4096, (float*)d_out, 4096, 4096LL,
         proj_b2, nullptr, 0, 0, 1, 4096, 256, 0);
}